// PDAGNN_36249523978837
// MI455X (gfx1250) — compile-verified
//
#include <hip/hip_runtime.h>
#include <hip/hip_bf16.h>

namespace {
constexpr int DDIM     = 64;   // feature dim
constexpr int NBJ      = 4;    // nodes per batch
constexpr int TILE_B   = 32;   // batches per block (500000 % 32 == 0)
constexpr int NTHREADS = 64;   // 2 waves (wave32)
constexpr int ST       = 68;   // padded LDS row stride in floats (bank-conflict free)

typedef __attribute__((ext_vector_type(2))) float v2f;
typedef __attribute__((ext_vector_type(4))) float v4f;
typedef __attribute__((ext_vector_type(8))) float v8f;

typedef unsigned int u32;
typedef __attribute__((ext_vector_type(4))) u32 u32x4;
typedef __attribute__((ext_vector_type(8))) int i32x8;
typedef __attribute__((ext_vector_type(4))) int i32x4;
} // namespace

// TDM: DMA a 2D tile (rows x 64 floats, row stride 64 floats in memory) from
// global into LDS, inserting 4 dwords of padding after every 64 dwords so the
// LDS image has row stride ST=68 floats (bank-conflict-free layout).
// D# encoding per CDNA5 ISA.
__device__ __forceinline__ void tdm_load_2d_pad68(u32 ldsAddr, const void* gptr, u32 rows)
{
    const unsigned long long ga = (unsigned long long)(uintptr_t)gptr;
    u32x4 g0;
    g0.x = 1u;                                   // count=1 (valid user descriptor)
    g0.y = ldsAddr;                              // lds_addr [63:32]
    g0.z = (u32)ga;                              // global_addr lo  [95:64]
    g0.w = ((u32)(ga >> 32) & 0x01FFFFFFu)       // global_addr hi  [120:96]
           | 0x80000000u;                        // type=2 ("image") [127:126]=10b

    i32x8 g1;
    g1[0] = (int)((2u << 16)                     // data_size = 2 -> 4 bytes
                | (1u << 20)                     // pad_enable
                | (5u << 22)                     // pad_interval: 2^(5+1)=64 dwords
                | (3u << 25));                   // pad_amount: 3+1 = 4 dwords
    g1[1] = (int)(64u << 16);                    // tensor_dim0 = 64 (bits 79:48)
    g1[2] = (int)(rows << 16);                   // tensor_dim1 = rows (bits 111:80)
    g1[3] = (int)(64u << 16);                    // tile_dim0 = 64 (bits 127:112)
    g1[4] = (int)rows;                           // tile_dim1 = rows (bits 143:128)
    g1[5] = 64;                                  // tensor_dim0_stride = 64 (bits 207:160)
    g1[6] = 0;
    g1[7] = 0;

    const i32x4 gz4 = {0, 0, 0, 0};              // groups 2/3 unused (2-D tile)
    const i32x8 gz8 = {0, 0, 0, 0, 0, 0, 0, 0};  // trailing group (6-arg form), unused
    __builtin_amdgcn_tensor_load_to_lds(g0, g1, gz4, gz4, gz8, 0);
}

__global__ __launch_bounds__(NTHREADS)
void gat_fused_kernel(const float* __restrict__ x,
                      const float* __restrict__ attW,
                      const float* __restrict__ attA,
                      float* __restrict__ out)
{
    __shared__ __align__(16) float sW[DDIM * ST];            // attW, padded rows
    __shared__ __align__(16) float sX[TILE_B * NBJ * ST];    // x tile, padded rows
    __shared__ __align__(16) float sY[TILE_B * ST];          // y = w @ x[b]
    __shared__ __align__(16) float sA[2 * DDIM];             // attA (a1 | a2)
    __shared__ __align__(16) float sV[2 * DDIM];             // v1 = attW@a1 | v2 = attW@a2
    __shared__ __align__(16) float sLR[TILE_B * 2 * 4];      // left/right exchange

    const int  t     = threadIdx.x;
    const long tile0 = (long)blockIdx.x * TILE_B;            // first batch of tile

    // ---- 1) stage attW (64x64) and the x tile (128 rows x 64) via TDM ----
    if (t < 32) {                                            // wave 0 issues the DMAs
        tdm_load_2d_pad68((u32)(uintptr_t)&sW[0], attW, 64u);
        tdm_load_2d_pad68((u32)(uintptr_t)&sX[0],
                          x + tile0 * (NBJ * DDIM), (u32)(TILE_B * NBJ));
        __builtin_amdgcn_s_wait_tensorcnt(0);
    }
    sA[t]      = attA[t];
    sA[t + 64] = attA[t + 64];
    __syncthreads();

    // ---- 2) v1[t] = attW[t,:]·a1 ; v2[t] = attW[t,:]·a2 ----
    {
        const float* wr = sW + t * ST;
        float a1 = 0.f, a2 = 0.f;
#pragma unroll
        for (int c = 0; c < 16; ++c) {
            const v4f w  = *(const v4f*)(wr + c * 4);
            const v4f u1 = *(const v4f*)(sA + c * 4);
            const v4f u2 = *(const v4f*)(sA + 64 + c * 4);
            a1 += w.x * u1.x + w.y * u1.y + w.z * u1.z + w.w * u1.w;
            a2 += w.x * u2.x + w.y * u2.y + w.z * u2.z + w.w * u2.w;
        }
        sV[t]      = a1;
        sV[64 + t] = a2;
    }
    __syncthreads();

    // ---- 3) attention: two threads per batch; thread handles rows i = 2p, 2p+1 ----
    const int m = t >> 1;        // local batch 0..31
    const int p = t & 1;
    float lf0 = 0.f, lf1 = 0.f, rt0 = 0.f, rt1 = 0.f;
#pragma unroll
    for (int ii = 0; ii < 2; ++ii) {
        const int i = 2 * p + ii;
        const float* xr = sX + (m * NBJ + i) * ST;
        float dl = 0.f, dr = 0.f;
#pragma unroll
        for (int c = 0; c < 16; ++c) {
            const v4f xv = *(const v4f*)(xr + c * 4);
            const v4f w1 = *(const v4f*)(sV + c * 4);
            const v4f w2 = *(const v4f*)(sV + 64 + c * 4);
            dl += xv.x * w1.x + xv.y * w1.y + xv.z * w1.z + xv.w * w1.w;
            dr += xv.x * w2.x + xv.y * w2.y + xv.z * w2.z + xv.w * w2.w;
        }
        if (ii == 0) { lf0 = dl; rt0 = dr; } else { lf1 = dl; rt1 = dr; }
    }
    {
        float* lr = sLR + (m * 2 + p) * 4;
        lr[0] = lf0; lr[1] = lf1; lr[2] = rt0; lr[3] = rt1;
    }
    __syncthreads();

    float L[4], R[4];
    {
        const float* lro = sLR + (m * 2 + (1 - p)) * 4;
        const int q = 2 * (1 - p);
        L[2 * p] = lf0; L[2 * p + 1] = lf1;
        R[2 * p] = rt0; R[2 * p + 1] = rt1;
        L[q] = lro[0]; L[q + 1] = lro[1];
        R[q] = lro[2]; R[q + 1] = lro[3];
    }

    // scores -> leaky_relu -> row softmax -> column sums w[j]
    float wj0 = 0.f, wj1 = 0.f, wj2 = 0.f, wj3 = 0.f;
#pragma unroll
    for (int i = 0; i < 4; ++i) {
        float s0 = L[i] + R[0]; s0 = fmaxf(s0, 0.2f * s0);
        float s1 = L[i] + R[1]; s1 = fmaxf(s1, 0.2f * s1);
        float s2 = L[i] + R[2]; s2 = fmaxf(s2, 0.2f * s2);
        float s3 = L[i] + R[3]; s3 = fmaxf(s3, 0.2f * s3);
        const float mx = fmaxf(fmaxf(s0, s1), fmaxf(s2, s3));
        s0 = __expf(s0 - mx); s1 = __expf(s1 - mx);
        s2 = __expf(s2 - mx); s3 = __expf(s3 - mx);
        const float inv = 1.0f / (s0 + s1 + s2 + s3);
        wj0 += s0 * inv; wj1 += s1 * inv; wj2 += s2 * inv; wj3 += s3 * inv;
    }

    // ---- 4) y[m,:] = sum_j w[j] * x[m,j,:]  (thread covers half the dims) ----
    {
        const float* xm = sX + m * NBJ * ST;
        float* ym = sY + m * ST;
#pragma unroll
        for (int c = 0; c < 8; ++c) {
            const int d = p * 32 + c * 4;
            const v4f x0 = *(const v4f*)(xm + 0 * ST + d);
            const v4f x1 = *(const v4f*)(xm + 1 * ST + d);
            const v4f x2 = *(const v4f*)(xm + 2 * ST + d);
            const v4f x3 = *(const v4f*)(xm + 3 * ST + d);
            const v4f acc = x0 * wj0 + x1 * wj1 + x2 * wj2 + x3 * wj3;
            *(v4f*)(ym + d) = acc;
        }
    }
    __syncthreads();

    // ---- 5) out[tile] = Y(32x64) @ attW(64x64) via V_WMMA_F32_16X16X4_F32 ----
    // A 16x4 f32 layout: lane (l<16) M=l holds K=2h+{0,1} in 2 VGPRs (h = l>>4)
    // C 16x16 f32 layout: vgpr r -> M = r + 8h, N = lane&15
    const int lane = t & 31;
    const int wv   = t >> 5;              // wave: 16 batches each
    const int col  = lane & 15;           // A-row (M) for A-frag / N-col for B/C
    const int h    = lane >> 4;
    const float* ya = sY + (wv * 16 + col) * ST + 2 * h;
    float* ob = out + (tile0 + wv * 16 + 8 * h) * DDIM + col;

#pragma unroll
    for (int n = 0; n < 4; ++n) {
        v8f acc = {0.f, 0.f, 0.f, 0.f, 0.f, 0.f, 0.f, 0.f};
#pragma unroll
        for (int k = 0; k < 16; ++k) {
            const v2f a = *(const v2f*)(ya + 4 * k);        // y[M, 4k+2h .. +1]
            v2f b;
            b.x = sW[(4 * k + 2 * h + 0) * ST + 16 * n + col];
            b.y = sW[(4 * k + 2 * h + 1) * ST + 16 * n + col];
            acc = __builtin_amdgcn_wmma_f32_16x16x4_f32(
                false, a, false, b, (short)0, acc, false, false);
        }
#pragma unroll
        for (int r = 0; r < 8; ++r)
            ob[(long)r * DDIM + 16 * n] = acc[r];
    }
}

extern "C" void kernel_launch(void* const* d_in, const int* in_sizes, int n_in,
                              void* d_out, int out_size, void* d_ws, size_t ws_size,
                              hipStream_t stream) {
    (void)n_in; (void)out_size; (void)d_ws; (void)ws_size;
    const float* x    = (const float*)d_in[0];
    const float* attW = (const float*)d_in[1];
    const float* attA = (const float*)d_in[2];
    float* out = (float*)d_out;

    const int nBatch = in_sizes[0] / (NBJ * DDIM);   // 500000
    const int grid   = nBatch / TILE_B;              // 15625 (exact)
    gat_fused_kernel<<<grid, NTHREADS, 0, stream>>>(x, attW, attA, out);
}